// RecurrentSpikingLayer_45380624450081
// MI455X (gfx1250) — compile-verified
//
#include <hip/hip_runtime.h>

// MI455X / gfx1250, wave32. Exact-fp32 path via V_WMMA_F32_16X16X4_F32,
// async global->LDS DMA (ASYNCcnt) for operand staging.
typedef __attribute__((ext_vector_type(2))) float v2f;
typedef __attribute__((ext_vector_type(8))) float v8f;

#define B_  128
#define T_  500
#define I_  700
#define N_  512
#define BTN (128L * 500L * 512L)

// ---------------------------------------------------------------------------
// Phase 1: h = inputs @ w   (M=B*T=64000, K=700, N=512), fp32 WMMA 16x16x4.
// h is stashed into the syn_rec region of d_out (slot t holds h[:,t]).
// Block = 32-row M panel (2000 blocks), 8 waves; wave computes 2 M-tiles x
// 4 N-tiles (8 accumulators) so each B fragment feeds two WMMAs.
// A panel is staged via GLOBAL_LOAD_ASYNC_TO_LDS_B128, double-buffered in
// K-chunks of 100 so the DMA of chunk c+1 overlaps compute of chunk c.
// ---------------------------------------------------------------------------
#define MBLK 32
#define KC   100
#define SPAD 108                 // LDS row stride (floats): 108%64 spreads banks
#define BUFF (MBLK * SPAD)       // floats per K-chunk buffer

__global__ __launch_bounds__(256) void snn_ff_gemm(const float* __restrict__ inp,
                                                   const float* __restrict__ w,
                                                   float* __restrict__ hout) {
    __shared__ float As[2 * BUFF];
    const int m0   = blockIdx.x * MBLK;
    const int lane = threadIdx.x & 31;
    const int wv   = threadIdx.x >> 5;   // wave 0..7
    const int hi   = lane >> 4;          // lane half carries K+2 (A and B)
    const int colr = lane & 15;

    // Issue one K-chunk of the A panel (32 rows x 100 fp32) as async DMA
    // into LDS buffer b. 16B-aligned on both sides -> b128 transfers.
    auto issue = [&](int c, int b) {
        const int kc = c * KC;
        for (int idx = threadIdx.x; idx < MBLK * (KC / 4); idx += 256) {
            const int row = idx / (KC / 4);
            const int c4  = idx % (KC / 4);
            const unsigned lds =
                (unsigned)(size_t)&As[b * BUFF + row * SPAD + c4 * 4];
            const unsigned long long ga = (unsigned long long)(size_t)(
                inp + (long)(m0 + row) * I_ + kc + c4 * 4);
            asm volatile("global_load_async_to_lds_b128 %0, %1, off"
                         :: "v"(lds), "v"(ga) : "memory");
        }
    };

    v8f acc[2][4];
#pragma unroll
    for (int i = 0; i < 2; ++i)
#pragma unroll
        for (int j = 0; j < 4; ++j)
#pragma unroll
            for (int r = 0; r < 8; ++r) acc[i][j][r] = 0.0f;

    // Per-lane B base: w[(k + 2*hi)*N + col]
    const float* pb[4];
#pragma unroll
    for (int j = 0; j < 4; ++j)
        pb[j] = w + (long)(2 * hi) * N_ + (wv * 64 + j * 16 + colr);

    issue(0, 0);
    for (int c = 0; c < I_ / KC; ++c) {          // 7 chunks
        asm volatile("s_wait_asynccnt 0" ::: "memory");  // chunk c landed (this wave)
        __syncthreads();                          // ...landed for all waves; and all
                                                  // waves done reading buffer (c+1)&1
        if (c + 1 < I_ / KC) issue(c + 1, (c + 1) & 1);  // overlap DMA with compute
        const float* bufp = &As[(c & 1) * BUFF];
        const int kb = c * KC;
        for (int kk = 0; kk < KC; kk += 4) {
            const v2f a0 = *(const v2f*)(bufp + colr * SPAD + kk + 2 * hi);
            const v2f a1 = *(const v2f*)(bufp + (16 + colr) * SPAD + kk + 2 * hi);
#pragma unroll
            for (int j = 0; j < 4; ++j) {
                v2f b;
                b.x = pb[j][(long)(kb + kk) * N_];
                b.y = pb[j][(long)(kb + kk + 1) * N_];
                acc[0][j] = __builtin_amdgcn_wmma_f32_16x16x4_f32(
                    false, a0, false, b, (short)0, acc[0][j], false, false);
                acc[1][j] = __builtin_amdgcn_wmma_f32_16x16x4_f32(
                    false, a1, false, b, (short)0, acc[1][j], false, false);
            }
        }
    }

    // C layout: element r -> row m0 + i*16 + r + 8*hi, column n0 + (lane&15).
#pragma unroll
    for (int i = 0; i < 2; ++i)
#pragma unroll
        for (int j = 0; j < 4; ++j) {
            const int n = wv * 64 + j * 16 + colr;
#pragma unroll
            for (int r = 0; r < 8; ++r)
                hout[(long)(m0 + i * 16 + r + 8 * hi) * N_ + n] = acc[i][j][r];
        }
}

// ---------------------------------------------------------------------------
// Phase 2: the recurrent scan. 8 blocks x 512 threads (16 waves).
// Block owns 16 batch rows, all 512 neurons. Wave w owns n-tiles w*32, w*32+16.
// Per step: spikes -> LDS (16x512), barrier, new_syn = alpha*syn + h_t + out@v
// via fp32 WMMA over K=512, elementwise mem update, store slot t+1.
// h[:,t] lives in the syn_rec slot t and is prefetched one step ahead before
// the slot is overwritten with new_syn (guarded by s_wait_loadcnt 0).
// ---------------------------------------------------------------------------
#define SPK_PAD 516

__global__ __launch_bounds__(512) void snn_scan(const float* __restrict__ v,
                                                const float* __restrict__ alpha,
                                                const float* __restrict__ beta,
                                                const float* __restrict__ th,
                                                const float* __restrict__ reset,
                                                const float* __restrict__ rest,
                                                float* __restrict__ out) {
    __shared__ float spkLds[16 * SPK_PAD];

    float* syn_r = out;                 // also h staging
    float* mem_r = out + BTN;
    float* spk_r = out + 2 * BTN;

    const int b0   = blockIdx.x * 16;
    const int lane = threadIdx.x & 31;
    const int wv   = threadIdx.x >> 5;  // 0..15
    const int hi   = lane >> 4;
    const int colr = lane & 15;

    // Warm L2 with the full v matrix (1 MB, re-read 499x by every block).
    {
        const char* vb = (const char*)v;
#pragma unroll
        for (int i = 0; i < 16; ++i)
            __builtin_prefetch(vb + (size_t)threadIdx.x * 2048 + (size_t)i * 128, 0, 1);
    }

    // Per-lane, per-tile neuron parameters (column = lane&15 in C layout).
    float al[2], be[2], thv[2], ombeta[2], omb_rest[2], thmr[2];
    int col[2];
#pragma unroll
    for (int j = 0; j < 2; ++j) {
        col[j]      = wv * 32 + j * 16 + colr;
        al[j]       = alpha[col[j]];
        be[j]       = beta[col[j]];
        thv[j]      = th[col[j]];
        ombeta[j]   = 1.0f - be[j];
        omb_rest[j] = ombeta[j] * rest[col[j]];
        thmr[j]     = thv[j] - reset[col[j]];
    }

    const long RSTRIDE = (long)T_ * N_;              // element r -> batch row r+8*hi
    const long base    = (long)(b0 + 8 * hi) * RSTRIDE;
    float *ps[2], *pm[2], *pk[2];
    const float* pv[2];
#pragma unroll
    for (int j = 0; j < 2; ++j) {
        ps[j] = syn_r + base + col[j];
        pm[j] = mem_r + base + col[j];
        pk[j] = spk_r + base + col[j];
        pv[j] = v + (long)(2 * hi) * N_ + col[j];    // v[(k+2*hi)*N + col]
    }

    v8f synv[2], memv[2], hc[2];
#pragma unroll
    for (int j = 0; j < 2; ++j)
#pragma unroll
        for (int r = 0; r < 8; ++r) { synv[j][r] = 0.0f; memv[j][r] = 0.0f; }

    // h[:,0] from slot 0, then overwrite t=0 slots with the initial zeros.
#pragma unroll
    for (int j = 0; j < 2; ++j)
#pragma unroll
        for (int r = 0; r < 8; ++r) hc[j][r] = ps[j][r * RSTRIDE];
    asm volatile("s_wait_loadcnt 0" ::: "memory");
#pragma unroll
    for (int j = 0; j < 2; ++j)
#pragma unroll
        for (int r = 0; r < 8; ++r) {
            ps[j][r * RSTRIDE] = 0.0f;
            pm[j][r * RSTRIDE] = 0.0f;
            pk[j][r * RSTRIDE] = 0.0f;
        }

    for (int t = 0; t < T_ - 1; ++t) {
        // 1) spikes from current mem; share via LDS (row = r+8*hi, col).
        v8f o[2];
#pragma unroll
        for (int j = 0; j < 2; ++j)
#pragma unroll
            for (int r = 0; r < 8; ++r) {
                const float s = (memv[j][r] - thv[j]) > 0.0f ? 1.0f : 0.0f;
                o[j][r] = s;
                spkLds[(r + 8 * hi) * SPK_PAD + col[j]] = s;
            }

        // 2) prefetch h[:,t+1] before its slot is clobbered below.
        v8f hn[2];
#pragma unroll
        for (int j = 0; j < 2; ++j)
#pragma unroll
            for (int r = 0; r < 8; ++r) hn[j][r] = ps[j][r * RSTRIDE + N_];

        __syncthreads();

        // 3) new_syn = alpha*syn + h_t + out @ v   (fp32 WMMA over K=512)
        v8f ns[2];
#pragma unroll
        for (int j = 0; j < 2; ++j)
#pragma unroll
            for (int r = 0; r < 8; ++r) ns[j][r] = al[j] * synv[j][r] + hc[j][r];

#pragma unroll 4
        for (int k = 0; k < N_; k += 4) {
            const v2f a = *(const v2f*)(&spkLds[colr * SPK_PAD + k + 2 * hi]);
#pragma unroll
            for (int j = 0; j < 2; ++j) {
                v2f b;
                b.x = pv[j][(long)k * N_];
                b.y = pv[j][(long)(k + 1) * N_];
                ns[j] = __builtin_amdgcn_wmma_f32_16x16x4_f32(
                    false, a, false, b, (short)0, ns[j], false, false);
            }
        }

        // 4) new_mem = beta*mem + (1-beta)*rest + (1-beta)*syn - out*(th-reset)
        v8f nm[2];
#pragma unroll
        for (int j = 0; j < 2; ++j)
#pragma unroll
            for (int r = 0; r < 8; ++r)
                nm[j][r] = be[j] * memv[j][r] + omb_rest[j] +
                           ombeta[j] * synv[j][r] - o[j][r] * thmr[j];

        // 5) h prefetch must have landed before new_syn overwrites slot t+1.
        asm volatile("s_wait_loadcnt 0" ::: "memory");
#pragma unroll
        for (int j = 0; j < 2; ++j)
#pragma unroll
            for (int r = 0; r < 8; ++r) {
                ps[j][r * RSTRIDE + N_] = ns[j][r];
                pm[j][r * RSTRIDE + N_] = nm[j][r];
                pk[j][r * RSTRIDE + N_] = o[j][r];
            }

        __syncthreads();   // protect LDS spikes before next step rewrites them

#pragma unroll
        for (int j = 0; j < 2; ++j) {
            synv[j] = ns[j];
            memv[j] = nm[j];
            hc[j]   = hn[j];
            ps[j] += N_; pm[j] += N_; pk[j] += N_;
        }
    }
}

extern "C" void kernel_launch(void* const* d_in, const int* in_sizes, int n_in,
                              void* d_out, int out_size, void* d_ws, size_t ws_size,
                              hipStream_t stream) {
    const float* inp   = (const float*)d_in[0];
    const float* w     = (const float*)d_in[1];
    const float* v     = (const float*)d_in[2];
    const float* alpha = (const float*)d_in[3];
    const float* beta  = (const float*)d_in[4];
    const float* th    = (const float*)d_in[5];
    const float* reset = (const float*)d_in[6];
    const float* rest  = (const float*)d_in[7];
    float* out = (float*)d_out;

    // Phase 1: h -> syn_rec region of d_out (slot t = h[:,t]).
    snn_ff_gemm<<<(B_ * T_) / MBLK, 256, 0, stream>>>(inp, w, out);
    // Phase 2: sequential scan, 8 blocks of 16 batch rows.
    snn_scan<<<B_ / 16, 512, 0, stream>>>(v, alpha, beta, th, reset, rest, out);
}